// LoraLinear_41403484733496
// MI455X (gfx1250) — compile-verified
//
#include <hip/hip_runtime.h>

// Multi-adapter LoRA on gfx1250 via grouped GEMM + V_WMMA_F32_16X16X4_F32.
// Memory-bound problem (~388 MB moved, ~2.1 GFLOP fp32): group tokens by
// adapter so compute stays at the useful FLOP count, keep fp32 end-to-end.

#define T_TOK 8192
#define H_DIM 4096
#define O_DIM 4096
#define R_RNK 16
#define N_AD  8
#define TILES (T_TOK / 16) // 512 row tiles

typedef __attribute__((ext_vector_type(2))) float v2f;
typedef __attribute__((ext_vector_type(8))) float v8f;

// ---------------------------------------------------------------------------
// Kernel 1: bucket tokens by adapter (histogram -> offsets -> scatter).
// ---------------------------------------------------------------------------
__global__ __launch_bounds__(1024) void lora_group_kernel(
    const int* __restrict__ aidx, int* __restrict__ offs,
    int* __restrict__ cnts, int* __restrict__ toks)
{
  __shared__ int h[N_AD];
  __shared__ int cur[N_AD];
  const int tid = threadIdx.x;
  if (tid < N_AD) h[tid] = 0;
  __syncthreads();
  for (int t = tid; t < T_TOK; t += blockDim.x) atomicAdd(&h[aidx[t]], 1);
  __syncthreads();
  if (tid == 0) {
    int s = 0;
    for (int a = 0; a < N_AD; ++a) { offs[a] = s; cnts[a] = h[a]; cur[a] = s; s += h[a]; }
  }
  __syncthreads();
  for (int t = tid; t < T_TOK; t += blockDim.x) {
    const int a = aidx[t];
    const int p = atomicAdd(&cur[a], 1);
    toks[p] = t;
  }
}

// ---------------------------------------------------------------------------
// Kernel 2: shrink  v[t,0..15] = x[t,:] @ lora_a[a(t)]   (grouped, split-K)
// Block = one 16-token tile of one adapter; 8 waves each own a K-slice of 512
// and accumulate a 16x16 (M=tokens, N=R) partial with f32 WMMA; LDS reduce.
// ---------------------------------------------------------------------------
__global__ __launch_bounds__(256) void lora_shrink_kernel(
    const float* __restrict__ x, const float* __restrict__ lora_a,
    const int* __restrict__ offs, const int* __restrict__ cnts,
    const int* __restrict__ toks, float* __restrict__ v)
{
  __shared__ float red[8][256];

  const int a    = blockIdx.x / TILES;
  const int tile = blockIdx.x % TILES;
  const int cnt  = cnts[a];
  const int row0 = tile * 16;
  if (row0 >= cnt) return;                 // uniform early-exit (wasted blocks)
  const int off    = offs[a];
  const int nvalid = min(16, cnt - row0);

  const int tid  = threadIdx.x;
  const int lane = tid & 31;
  const int wave = tid >> 5;
  const int n    = lane & 15;              // M of A-frag row / N of B-frag col
  const int krow = (lane >> 4) * 2;        // K sub-offset within a K=4 step

  // Gathered x row for this lane's A-matrix row (clamp ragged rows to row 0;
  // their D rows are never stored).
  const int mrow = (n < nvalid) ? n : 0;
  const int tok  = toks[off + row0 + mrow];
  const float* __restrict__ xrow = x + (size_t)tok * H_DIM;
  const float* __restrict__ arow = lora_a + (size_t)a * (H_DIM * R_RNK);

  v8f acc = {};
  const int k0 = wave * (H_DIM / 8);
  const int k1 = k0 + (H_DIM / 8);
  for (int k = k0; k < k1; k += 4) {
    // A 16x4 f32 frag: lane holds (M = lane%16, K = k + krow + {0,1})
    const v2f af = *(const v2f*)(xrow + k + krow);
    // B 4x16 f32 frag: lane holds (K = k + krow + {0,1}, N = lane%16)
    v2f bf;
    bf.x = arow[(size_t)(k + krow)     * R_RNK + n];
    bf.y = arow[(size_t)(k + krow + 1) * R_RNK + n];
    acc = __builtin_amdgcn_wmma_f32_16x16x4_f32(false, af, false, bf,
                                                (short)0, acc, false, false);
  }

  #pragma unroll
  for (int j = 0; j < 8; ++j) red[wave][j * 32 + lane] = acc[j];
  __syncthreads();

  // 256 threads: each reduces one element of the 16x16 tile across 8 waves.
  float s = 0.f;
  #pragma unroll
  for (int w = 0; w < 8; ++w) s += red[w][tid];
  const int jj = tid >> 5;
  const int ll = tid & 31;
  const int M  = jj + 8 * (ll >> 4);       // C/D layout: VGPR j -> M=j / j+8
  const int N  = ll & 15;
  if (M < nvalid) {
    const int tm = toks[off + row0 + M];
    v[(size_t)tm * R_RNK + N] = s;
  }
}

// ---------------------------------------------------------------------------
// Kernel 3: expand  out[t,:] = result[t,:] + v[t,:] @ lora_b[a(t)]
// Block = one 16-token tile; 8 waves stride the 256 column tiles. result is
// loaded straight into the WMMA accumulator; 4 chained K=4 WMMAs cover R=16.
// ---------------------------------------------------------------------------
__global__ __launch_bounds__(256) void lora_expand_kernel(
    const float* __restrict__ result, const float* __restrict__ lora_b,
    const int* __restrict__ offs, const int* __restrict__ cnts,
    const int* __restrict__ toks, const float* __restrict__ v,
    float* __restrict__ out)
{
  const int a    = blockIdx.x / TILES;
  const int tile = blockIdx.x % TILES;
  const int cnt  = cnts[a];
  const int row0 = tile * 16;
  if (row0 >= cnt) return;
  const int off    = offs[a];
  const int nvalid = min(16, cnt - row0);

  const int tid  = threadIdx.x;
  const int lane = tid & 31;
  const int wave = tid >> 5;
  const int n    = lane & 15;
  const int half = lane >> 4;
  const int krow = half * 2;

  // A fragments from v (16 tokens x 16 R), one v2f per K=4 WMMA step.
  const int am = (n < nvalid) ? n : 0;
  const float* __restrict__ vrow = v + (size_t)toks[off + row0 + am] * R_RNK;
  v2f afr[4];
  #pragma unroll
  for (int q = 0; q < 4; ++q)
    afr[q] = *(const v2f*)(vrow + q * 4 + krow);

  // Per-lane gathered row bases for the 8 C/D rows this lane holds.
  const int mbase = half * 8;
  size_t rb[8];
  #pragma unroll
  for (int j = 0; j < 8; ++j) {
    const int M  = j + mbase;
    const int mr = (M < nvalid) ? M : 0;
    rb[j] = (size_t)toks[off + row0 + mr] * O_DIM;
  }

  const float* __restrict__ brow = lora_b + (size_t)a * (R_RNK * O_DIM);

  for (int ct = wave; ct < O_DIM / 16; ct += 8) {
    const int col = ct * 16 + n;
    v8f c;
    #pragma unroll
    for (int j = 0; j < 8; ++j) c[j] = result[rb[j] + col];
    #pragma unroll
    for (int q = 0; q < 4; ++q) {
      v2f bf;
      bf.x = brow[(size_t)(q * 4 + krow)     * O_DIM + col];
      bf.y = brow[(size_t)(q * 4 + krow + 1) * O_DIM + col];
      c = __builtin_amdgcn_wmma_f32_16x16x4_f32(false, afr[q], false, bf,
                                                (short)0, c, false, false);
    }
    #pragma unroll
    for (int j = 0; j < 8; ++j)
      if (j + mbase < nvalid) out[rb[j] + col] = c[j];
  }
}

// ---------------------------------------------------------------------------
extern "C" void kernel_launch(void* const* d_in, const int* in_sizes, int n_in,
                              void* d_out, int out_size, void* d_ws, size_t ws_size,
                              hipStream_t stream) {
  const float* result = (const float*)d_in[0];
  const float* x      = (const float*)d_in[1];
  const float* lora_a = (const float*)d_in[2];
  const float* lora_b = (const float*)d_in[3];
  const int*   aidx   = (const int*)d_in[4];
  float* out = (float*)d_out;

  // Workspace layout: offs[8] | cnts[8] | toks[8192] | v[8192*16] (~557 KB)
  int* offs = (int*)d_ws;
  int* cnts = offs + N_AD;
  int* toks = cnts + N_AD;
  float* v  = (float*)(toks + T_TOK);

  lora_group_kernel<<<1, 1024, 0, stream>>>(aidx, offs, cnts, toks);
  lora_shrink_kernel<<<N_AD * TILES, 256, 0, stream>>>(x, lora_a, offs, cnts, toks, v);
  lora_expand_kernel<<<N_AD * TILES, 256, 0, stream>>>(result, lora_b, offs, cnts, toks, v, out);
}